// TopkRouter_69114613727660
// MI455X (gfx1250) — compile-verified
//
#include <hip/hip_runtime.h>
#include <hip/hip_bf16.h>

typedef float v2f __attribute__((ext_vector_type(2)));
typedef float v8f __attribute__((ext_vector_type(8)));

#define DIM            1024     // D
#define NEXP           64       // E
#define WAVES_PER_WG   4
#define ROWS_PER_WAVE  32       // 2 M-tiles of 16
#define ROWS_PER_WG    (WAVES_PER_WG * ROWS_PER_WAVE)   // 128
#define LDS_STRIDE     65       // 64 + 1 pad -> conflict-free row reads

__launch_bounds__(ROWS_PER_WG)
__global__ void topk_router_kernel(const float* __restrict__ x,
                                   const float* __restrict__ W,
                                   const float* __restrict__ bias,
                                   float* __restrict__ out_probs,
                                   float* __restrict__ out_idx)
{
    __shared__ float lds_logits[ROWS_PER_WG * LDS_STRIDE];
    __shared__ float lds_p1[ROWS_PER_WG];
    __shared__ float lds_p2[ROWS_PER_WG];
    __shared__ int   lds_i1[ROWS_PER_WG];
    __shared__ int   lds_i2[ROWS_PER_WG];

    const int tid   = threadIdx.x;
    const int wave  = tid >> 5;          // wave32
    const int lane  = tid & 31;
    const int lhalf = lane & 15;         // N / M position within half-wave
    const int khalf = lane >> 4;         // selects K half (A and B layouts)

    const int wg_row0   = blockIdx.x * ROWS_PER_WG;
    const int wave_row0 = wg_row0 + wave * ROWS_PER_WAVE;

    // ---- Phase 1: logits = x @ W via V_WMMA_F32_16X16X4_F32 ----
    v8f acc[2][4];
#pragma unroll
    for (int mt = 0; mt < 2; ++mt)
#pragma unroll
        for (int nt = 0; nt < 4; ++nt)
            acc[mt][nt] = (v8f)(0.0f);

    // bias per lane for each N-tile (same for both lane halves)
    float bias_v[4];
#pragma unroll
    for (int nt = 0; nt < 4; ++nt)
        bias_v[nt] = bias[nt * 16 + lhalf];

    // A row pointers: lane holds row (wave_row0 + mtile*16 + lhalf)
    const float* a0 = x + (size_t)(wave_row0 +  0 + lhalf) * DIM;
    const float* a1 = x + (size_t)(wave_row0 + 16 + lhalf) * DIM;

    for (int kb = 0; kb < DIM; kb += 4) {
        // A layout (16x4 f32): K = khalf*2 + vgpr  -> contiguous float2 per lane
        const int ka = kb + khalf * 2;
        v2f A0 = *(const v2f*)(a0 + ka);
        v2f A1 = *(const v2f*)(a1 + ka);

        // B layout (4x16 f32): K = khalf*2 + vgpr, N = lhalf
        const float* wp = W + (size_t)(kb + khalf * 2) * NEXP + lhalf;
#pragma unroll
        for (int nt = 0; nt < 4; ++nt) {
            v2f B;
            B.x = wp[nt * 16];          // K = kb + khalf*2
            B.y = wp[nt * 16 + NEXP];   // K = kb + khalf*2 + 1
            acc[0][nt] = __builtin_amdgcn_wmma_f32_16x16x4_f32(
                false, A0, false, B, (short)0, acc[0][nt], false, false);
            acc[1][nt] = __builtin_amdgcn_wmma_f32_16x16x4_f32(
                false, A1, false, B, (short)0, acc[1][nt], false, false);
        }
    }

    // C/D layout: VGPR r, lanes 0-15 -> M=r, N=lane; lanes 16-31 -> M=8+r
#pragma unroll
    for (int mt = 0; mt < 2; ++mt) {
#pragma unroll
        for (int nt = 0; nt < 4; ++nt) {
#pragma unroll
            for (int r = 0; r < 8; ++r) {
                const int row_local = wave * ROWS_PER_WAVE + mt * 16 + khalf * 8 + r;
                const int col       = nt * 16 + lhalf;
                lds_logits[row_local * LDS_STRIDE + col] = acc[mt][nt][r] + bias_v[nt];
            }
        }
    }

    __syncthreads();

    // ---- Phase 2: per-row top-2 + 2-term softmax ----
    {
        const float* rowp = &lds_logits[tid * LDS_STRIDE];
        float m1 = -__builtin_huge_valf(), m2 = -__builtin_huge_valf();
        int   i1 = 0, i2 = 0;
#pragma unroll 8
        for (int j = 0; j < NEXP; ++j) {
            float v = rowp[j];
            if (v > m1) { m2 = m1; i2 = i1; m1 = v; i1 = j; }
            else if (v > m2) { m2 = v; i2 = j; }
        }
        const float e   = __expf(m2 - m1);
        const float inv = 1.0f / (1.0f + e);
        lds_p1[tid] = inv;          // exp(0) / (1 + e)
        lds_p2[tid] = e * inv;
        lds_i1[tid] = i1;
        lds_i2[tid] = i2;
    }

    __syncthreads();

    // ---- Phase 3: coalesced scatter of sparse probs + indices ----
    float* probs_base = out_probs + (size_t)wg_row0 * NEXP;
#pragma unroll 4
    for (int it = 0; it < (ROWS_PER_WG * NEXP) / ROWS_PER_WG; ++it) {
        const int flat = it * ROWS_PER_WG + tid;
        const int r    = flat >> 6;       // /64
        const int c    = flat & (NEXP - 1);
        float v = 0.0f;
        if (c == lds_i1[r])      v = lds_p1[r];
        else if (c == lds_i2[r]) v = lds_p2[r];
        probs_base[flat] = v;
    }

    // indices (concatenated after probs; stored as float per output promotion)
    out_idx[(size_t)(wg_row0 + tid) * 2 + 0] = (float)lds_i1[tid];
    out_idx[(size_t)(wg_row0 + tid) * 2 + 1] = (float)lds_i2[tid];
}

extern "C" void kernel_launch(void* const* d_in, const int* in_sizes, int n_in,
                              void* d_out, int out_size, void* d_ws, size_t ws_size,
                              hipStream_t stream) {
    const float* x = (const float*)d_in[0];   // [N, D]
    const float* W = (const float*)d_in[1];   // [D, E]
    const float* b = (const float*)d_in[2];   // [E]
    // d_in[3], d_in[4] (noise params) unused: training=False path

    const int n_tok = in_sizes[0] / DIM;      // 32768

    float* probs = (float*)d_out;                       // [N, E]
    float* idx   = probs + (size_t)n_tok * NEXP;        // [N, 2] as floats

    dim3 grid(n_tok / ROWS_PER_WG);
    dim3 block(ROWS_PER_WG);
    hipLaunchKernelGGL(topk_router_kernel, grid, block, 0, stream,
                       x, W, b, probs, idx);
}